// GAT_16578573763128
// MI455X (gfx1250) — compile-verified
//
#include <hip/hip_runtime.h>
#include <hip/hip_bf16.h>
#include <math.h>

#define GAT_HEADS 8
#define GAT_HID   16
#define GAT_INC   128
#define GAT_OUTC  64

typedef float v2f __attribute__((ext_vector_type(2)));
typedef float v8f __attribute__((ext_vector_type(8)));

__device__ __forceinline__ float gat_lrelu(float x) { return x > 0.f ? x : 0.2f * x; }

// Order-independent float atomic max via monotone integer atomics.
__device__ __forceinline__ void gat_atomicMaxF(float* addr, float v) {
    if (v >= 0.f) atomicMax((int*)addr, __float_as_int(v));
    else          atomicMin((unsigned int*)addr, __float_as_uint(v));
}

// ---------------------------------------------------------------- utilities
__global__ void gat_fill_kernel(float* __restrict__ p, float v, int n) {
    int i = blockIdx.x * blockDim.x + threadIdx.x;
    if (i < n) p[i] = v;
}

__global__ void gat_init_out_kernel(float* __restrict__ out, const float* __restrict__ b2, int total) {
    int i = blockIdx.x * blockDim.x + threadIdx.x;
    if (i < total) out[i] = b2[i & (GAT_OUTC - 1)];
}

// ---------------------------------------------------------------- fp32 WMMA GEMM
// C[nrows x NCOL] = A[nrows x 128] @ W[128 x NCOL], all row-major fp32.
// One wave per 16-row strip; W staged in LDS in B-fragment order so each lane's
// fragment is a single aligned ds_load_b64; V_WMMA_F32_16X16X4_F32, K-loop of 32.
template <int NCOL>
__global__ __launch_bounds__(128) void gat_gemm_wmma_f32(const float* __restrict__ A,
                                                         const float* __restrict__ W,
                                                         float* __restrict__ C, int nrows) {
    // sW2[p*NCOL + col] = (W[2p][col], W[2p+1][col]) : the two K-values lane needs.
    __shared__ float2 sW2[64 * NCOL];
    for (int idx = threadIdx.x; idx < 64 * NCOL; idx += 128) {
        int p = idx / NCOL, col = idx - p * NCOL;
        float2 w;
        w.x = W[(2 * p) * NCOL + col];
        w.y = W[(2 * p + 1) * NCOL + col];
        sW2[idx] = w;
    }
    __syncthreads();

    const int wave = threadIdx.x >> 5;
    const int lane = threadIdx.x & 31;
    const int strip = blockIdx.x * 4 + wave;
    const int nstrips = (nrows + 15) >> 4;
    if (strip >= nstrips) return;           // wave-uniform: EXEC stays all-1s for WMMA

    const int hi = lane >> 4;               // 0 | 1
    const int lo = lane & 15;
    const int koff = hi * 2;
    int row = strip * 16 + lo;
    if (row >= nrows) row = nrows - 1;      // clamp for safety (N is a multiple of 16 here)

    v8f acc[NCOL / 16];
#pragma unroll
    for (int j = 0; j < NCOL / 16; ++j)
#pragma unroll
        for (int v = 0; v < 8; ++v) acc[j][v] = 0.f;

    const float2* ap2 = (const float2*)(A + (size_t)row * 128);
    for (int k0 = 0; k0 < 128; k0 += 4) {
        // A 16x4 fragment: lanes 0-15 rows M, K = k0+2*hi, k0+2*hi+1 (ISA 7.12.2)
        float2 af = ap2[(k0 + koff) >> 1];
        v2f a; a.x = af.x; a.y = af.y;
        const int p = (k0 + koff) >> 1;     // pair row in fragment-ordered LDS
#pragma unroll
        for (int j = 0; j < NCOL / 16; ++j) {
            float2 bf = sW2[p * NCOL + j * 16 + lo];   // single ds_load_b64
            v2f b; b.x = bf.x; b.y = bf.y;
            acc[j] = __builtin_amdgcn_wmma_f32_16x16x4_f32(
                false, a, false, b, (short)0, acc[j], false, false);
        }
    }

#pragma unroll
    for (int j = 0; j < NCOL / 16; ++j)
#pragma unroll
        for (int v = 0; v < 8; ++v) {
            int r = strip * 16 + v + hi * 8;   // C/D: VGPR v -> M = v + 8*hi, N = lo
            if (r < nrows) C[(size_t)r * NCOL + j * 16 + lo] = acc[j][v];
        }
}

// ---------------------------------------------------------------- layer 1 attention
__global__ void gat_attn1_kernel(const float* __restrict__ h1,
                                 const float* __restrict__ a_src, const float* __restrict__ a_dst,
                                 float* __restrict__ als, float* __restrict__ ald, int total) {
    int i = blockIdx.x * blockDim.x + threadIdx.x;  // over N*8
    if (i >= total) return;
    int n = i >> 3, h = i & 7;
    const float* hp = h1 + (size_t)n * 128 + h * 16;
    float s = 0.f, d = 0.f;
#pragma unroll
    for (int c = 0; c < 16; ++c) {
        float v = hp[c];
        s += v * a_src[h * 16 + c];
        d += v * a_dst[h * 16 + c];
    }
    als[i] = s;
    ald[i] = d;
}

__global__ void gat_edge_max1_kernel(const int* __restrict__ src, const int* __restrict__ dst,
                                     const float* __restrict__ als, const float* __restrict__ ald,
                                     float* __restrict__ m, int total) {
    int i = blockIdx.x * blockDim.x + threadIdx.x;  // over E*8
    if (i >= total) return;
    int e = i >> 3, h = i & 7;
    int s = src[e], t = dst[e];
    float x = gat_lrelu(als[s * 8 + h] + ald[t * 8 + h]);
    gat_atomicMaxF(&m[t * 8 + h], x);
}

__global__ void gat_edge_sum1_kernel(const int* __restrict__ src, const int* __restrict__ dst,
                                     const float* __restrict__ als, const float* __restrict__ ald,
                                     const float* __restrict__ m, float* __restrict__ den, int total) {
    int i = blockIdx.x * blockDim.x + threadIdx.x;
    if (i >= total) return;
    int e = i >> 3, h = i & 7;
    int s = src[e], t = dst[e];
    float x = gat_lrelu(als[s * 8 + h] + ald[t * 8 + h]);
    atomicAdd(&den[t * 8 + h], expf(x - m[t * 8 + h]));
}

__global__ void gat_edge_aggr1_kernel(const int* __restrict__ src, const int* __restrict__ dst,
                                      const float* __restrict__ als, const float* __restrict__ ald,
                                      const float* __restrict__ m, const float* __restrict__ den,
                                      const float* __restrict__ h1, float* __restrict__ aggr, int total) {
    int i = blockIdx.x * blockDim.x + threadIdx.x;  // over E*8: one (edge, head) per thread
    if (i >= total) return;
    int e = i >> 3, h = i & 7;
    int s = src[e], t = dst[e];
    float x = gat_lrelu(als[s * 8 + h] + ald[t * 8 + h]);
    float alpha = expf(x - m[t * 8 + h]) / (den[t * 8 + h] + 1e-16f);
    const float4* hp = (const float4*)(h1 + (size_t)s * 128 + h * 16);
    float* op = aggr + (size_t)t * 128 + h * 16;
#pragma unroll
    for (int q = 0; q < 4; ++q) {
        float4 v = hp[q];
        atomicAdd(op + q * 4 + 0, alpha * v.x);
        atomicAdd(op + q * 4 + 1, alpha * v.y);
        atomicAdd(op + q * 4 + 2, alpha * v.z);
        atomicAdd(op + q * 4 + 3, alpha * v.w);
    }
}

__global__ void gat_elu_bias_kernel(const float* __restrict__ aggr, const float* __restrict__ b,
                                    float* __restrict__ out, int total) {
    int i = blockIdx.x * blockDim.x + threadIdx.x;  // over N*128
    if (i >= total) return;
    float v = aggr[i] + b[i & 127];
    out[i] = v > 0.f ? v : expm1f(v);
}

// ---------------------------------------------------------------- layer 2 attention
__global__ void gat_attn2_kernel(const float* __restrict__ g,
                                 const float* __restrict__ a_src, const float* __restrict__ a_dst,
                                 float* __restrict__ als, float* __restrict__ ald, int n) {
    int i = blockIdx.x * blockDim.x + threadIdx.x;  // over N
    if (i >= n) return;
    const float* gp = g + (size_t)i * 64;
    float s = 0.f, d = 0.f;
#pragma unroll 8
    for (int c = 0; c < 64; ++c) {
        float v = gp[c];
        s += v * a_src[c];
        d += v * a_dst[c];
    }
    als[i] = s;
    ald[i] = d;
}

__global__ void gat_edge_max2_kernel(const int* __restrict__ src, const int* __restrict__ dst,
                                     const float* __restrict__ als, const float* __restrict__ ald,
                                     float* __restrict__ m, int total) {
    int e = blockIdx.x * blockDim.x + threadIdx.x;  // over E
    if (e >= total) return;
    int s = src[e], t = dst[e];
    float x = gat_lrelu(als[s] + ald[t]);
    gat_atomicMaxF(&m[t], x);
}

__global__ void gat_edge_sum2_kernel(const int* __restrict__ src, const int* __restrict__ dst,
                                     const float* __restrict__ als, const float* __restrict__ ald,
                                     const float* __restrict__ m, float* __restrict__ den, int total) {
    int e = blockIdx.x * blockDim.x + threadIdx.x;
    if (e >= total) return;
    int s = src[e], t = dst[e];
    float x = gat_lrelu(als[s] + ald[t]);
    atomicAdd(&den[t], expf(x - m[t]));
}

__global__ void gat_edge_aggr2_kernel(const int* __restrict__ src, const int* __restrict__ dst,
                                      const float* __restrict__ als, const float* __restrict__ ald,
                                      const float* __restrict__ m, const float* __restrict__ den,
                                      const float* __restrict__ g, float* __restrict__ out, int total) {
    int i = blockIdx.x * blockDim.x + threadIdx.x;  // over E*4: 16 channels per thread
    if (i >= total) return;
    int e = i >> 2, q = i & 3;
    int s = src[e], t = dst[e];
    float x = gat_lrelu(als[s] + ald[t]);
    float alpha = expf(x - m[t]) / (den[t] + 1e-16f);
    const float4* gp = (const float4*)(g + (size_t)s * 64 + q * 16);
    float* op = out + (size_t)t * 64 + q * 16;
#pragma unroll
    for (int r = 0; r < 4; ++r) {
        float4 v = gp[r];
        atomicAdd(op + r * 4 + 0, alpha * v.x);
        atomicAdd(op + r * 4 + 1, alpha * v.y);
        atomicAdd(op + r * 4 + 2, alpha * v.z);
        atomicAdd(op + r * 4 + 3, alpha * v.w);
    }
}

// ---------------------------------------------------------------- driver
extern "C" void kernel_launch(void* const* d_in, const int* in_sizes, int n_in,
                              void* d_out, int out_size, void* d_ws, size_t ws_size,
                              hipStream_t stream) {
    const float* x   = (const float*)d_in[0];
    const int*   ei  = (const int*)d_in[1];
    const float* W1  = (const float*)d_in[2];
    const float* as1 = (const float*)d_in[3];
    const float* ad1 = (const float*)d_in[4];
    const float* b1  = (const float*)d_in[5];
    const float* W2  = (const float*)d_in[6];
    const float* as2 = (const float*)d_in[7];
    const float* ad2 = (const float*)d_in[8];
    const float* b2  = (const float*)d_in[9];
    float* out = (float*)d_out;

    const int N = in_sizes[0] / 128;
    const int E = in_sizes[1] / 2;
    const int* src = ei;
    const int* dst = ei + E;

    // workspace layout (floats)
    float* ws    = (float*)d_ws;
    float* h1    = ws;                         // N*128 (reused as h2 after ELU)
    float* aggr1 = h1 + (size_t)N * 128;       // N*128
    float* g     = aggr1 + (size_t)N * 128;    // N*64
    float* als1  = g + (size_t)N * 64;         // N*8
    float* ald1  = als1 + (size_t)N * 8;       // N*8
    float* m1    = ald1 + (size_t)N * 8;       // N*8
    float* d1    = m1 + (size_t)N * 8;         // N*8
    float* als2  = d1 + (size_t)N * 8;         // N
    float* ald2  = als2 + (size_t)N;           // N
    float* m2    = ald2 + (size_t)N;           // N
    float* d2    = m2 + (size_t)N;             // N
    (void)ws_size; (void)n_in; (void)out_size;

    const int TB = 256;
    auto cdiv = [](long a, long b) { return (int)((a + b - 1) / b); };
    const float NEG_INF = -__builtin_huge_valf();
    const int nstrips = (N + 15) / 16;

    // ---- layer 1: h1 = x @ W1 (fp32 WMMA)
    gat_gemm_wmma_f32<128><<<cdiv(nstrips, 4), 128, 0, stream>>>(x, W1, h1, N);
    gat_attn1_kernel<<<cdiv((long)N * 8, TB), TB, 0, stream>>>(h1, as1, ad1, als1, ald1, N * 8);
    gat_fill_kernel<<<cdiv((long)N * 8, TB), TB, 0, stream>>>(m1, NEG_INF, N * 8);
    gat_fill_kernel<<<cdiv((long)N * 8, TB), TB, 0, stream>>>(d1, 0.f, N * 8);
    gat_fill_kernel<<<cdiv((long)N * 128, TB), TB, 0, stream>>>(aggr1, 0.f, N * 128);
    gat_edge_max1_kernel<<<cdiv((long)E * 8, TB), TB, 0, stream>>>(src, dst, als1, ald1, m1, E * 8);
    gat_edge_sum1_kernel<<<cdiv((long)E * 8, TB), TB, 0, stream>>>(src, dst, als1, ald1, m1, d1, E * 8);
    gat_edge_aggr1_kernel<<<cdiv((long)E * 8, TB), TB, 0, stream>>>(src, dst, als1, ald1, m1, d1,
                                                                    h1, aggr1, E * 8);
    // h2 = elu(aggr1 + b1), written over h1's buffer
    gat_elu_bias_kernel<<<cdiv((long)N * 128, TB), TB, 0, stream>>>(aggr1, b1, h1, N * 128);

    // ---- layer 2: g = h2 @ W2 (fp32 WMMA)
    gat_gemm_wmma_f32<64><<<cdiv(nstrips, 4), 128, 0, stream>>>(h1, W2, g, N);
    gat_attn2_kernel<<<cdiv((long)N, TB), TB, 0, stream>>>(g, as2, ad2, als2, ald2, N);
    gat_fill_kernel<<<cdiv((long)N, TB), TB, 0, stream>>>(m2, NEG_INF, N);
    gat_fill_kernel<<<cdiv((long)N, TB), TB, 0, stream>>>(d2, 0.f, N);
    gat_init_out_kernel<<<cdiv((long)N * 64, TB), TB, 0, stream>>>(out, b2, N * 64);
    gat_edge_max2_kernel<<<cdiv((long)E, TB), TB, 0, stream>>>(src, dst, als2, ald2, m2, E);
    gat_edge_sum2_kernel<<<cdiv((long)E, TB), TB, 0, stream>>>(src, dst, als2, ald2, m2, d2, E);
    gat_edge_aggr2_kernel<<<cdiv((long)E * 4, TB), TB, 0, stream>>>(src, dst, als2, ald2, m2, d2,
                                                                    g, out, E * 4);
}